// SelfAttentionLayer_17824114278642
// MI455X (gfx1250) — compile-verified
//
#include <hip/hip_runtime.h>
#include <hip/hip_bf16.h>
#include <math.h>

// Self-attention (B=4, S=4096, D=32, fp32) for gfx1250 using
// V_WMMA_F32_16X16X4_F32 for all GEMMs, flash-attention (online softmax).
// v2: 32 keys per inner iteration (halves softmax VALU overhead per key),
//     V stored transposed [B][D][S] so P@V B-operands are b64 loads,
//     global_prefetch for next K/V tiles, 8 waves per block for WGP$ reuse.

#define B_ 4
#define S_ 4096
#define D_ 32

typedef float v2f __attribute__((ext_vector_type(2)));
typedef float v8f __attribute__((ext_vector_type(8)));

__device__ __forceinline__ v8f wmma_f32(v2f a, v2f b, v8f c) {
  // (neg_a, A, neg_b, B, c_mod, C, reuse_a, reuse_b)
  return __builtin_amdgcn_wmma_f32_16x16x4_f32(false, a, false, b, (short)0, c,
                                               false, false);
}

// ---------------------------------------------------------------------------
// Kernel 1: Q/K/V projections.  X[B*S,32] @ W[32,32] + b  (x3 weight sets)
// One wave per 16-row tile; Q,K stored row-major, V stored transposed
// as Vt[B][D][S] so the attention kernel's P@V B-operand is contiguous.
// ---------------------------------------------------------------------------
__global__ __launch_bounds__(256) void qkv_proj_kernel(
    const float* __restrict__ X,
    const float* __restrict__ Wq, const float* __restrict__ bq,
    const float* __restrict__ Wk, const float* __restrict__ bk,
    const float* __restrict__ Wv, const float* __restrict__ bv,
    float* __restrict__ Qo, float* __restrict__ Ko, float* __restrict__ Vt) {
  const int lane = threadIdx.x & 31;
  const int wave = threadIdx.x >> 5;
  const int tile = blockIdx.x * 8 + wave;  // 0..1023
  const int row0 = tile * 16;
  const int h = lane >> 4;  // lane half
  const int m = lane & 15;

  // A operand: rows of X.  A[m][K] lives at lane (h,m), vgpr i, K = 4k+2h+i
  v2f a[8];
  const float* xrow = X + (size_t)(row0 + m) * D_;
#pragma unroll
  for (int k = 0; k < 8; ++k) a[k] = *(const v2f*)(xrow + 4 * k + 2 * h);

  const float* Ws[3] = {Wq, Wk, Wv};
  const float* bs[3] = {bq, bk, bv};

  const int bidx = row0 >> 12;        // batch (row0 / 4096)
  const int sl = row0 & (S_ - 1);     // row within batch

#pragma unroll
  for (int mat = 0; mat < 3; ++mat) {
    const float* W = Ws[mat];
    const float* bias = bs[mat];
#pragma unroll
    for (int nt = 0; nt < 2; ++nt) {
      const int n = nt * 16 + m;
      v8f c;
      const float bb0 = bias[n];
#pragma unroll
      for (int r = 0; r < 8; ++r) c[r] = bb0;  // C init = bias broadcast
#pragma unroll
      for (int k = 0; k < 8; ++k) {
        const int f = 4 * k + 2 * h;
        v2f bb;
        bb.x = W[(f + 0) * D_ + n];
        bb.y = W[(f + 1) * D_ + n];
        c = wmma_f32(a[k], bb, c);
      }
      if (mat == 0) {
#pragma unroll
        for (int r = 0; r < 8; ++r)
          Qo[(size_t)(row0 + r + 8 * h) * D_ + n] = c[r];
      } else if (mat == 1) {
#pragma unroll
        for (int r = 0; r < 8; ++r)
          Ko[(size_t)(row0 + r + 8 * h) * D_ + n] = c[r];
      } else {
        // V transposed: Vt[(b*D + n)*S + s]
#pragma unroll
        for (int r = 0; r < 8; ++r)
          Vt[((size_t)bidx * D_ + n) * S_ + sl + r + 8 * h] = c[r];
      }
    }
  }
}

// ---------------------------------------------------------------------------
// Kernel 2: fused flash attention + final projection + tanh.
// One wave per (batch, 16-query-row tile); 128 iterations of 32 keys each.
// ---------------------------------------------------------------------------
__global__ __launch_bounds__(256) void attn_kernel(
    const float* __restrict__ Q, const float* __restrict__ Kmat,
    const float* __restrict__ Vt, const float* __restrict__ Wf,
    const float* __restrict__ bf, float* __restrict__ out) {
  __shared__ float lds[8][16 * 32];  // per-wave staging (P tile / ctx tile)

  const int lane = threadIdx.x & 31;
  const int wave = threadIdx.x >> 5;
  const int tile = blockIdx.x * 8 + wave;  // 0..1023
  const int b = tile >> 8;                 // 256 query tiles per batch
  const int qt = tile & 255;
  const int row0 = b * S_ + qt * 16;       // row into flattened [B*S,32]
  const int h = lane >> 4;
  const int m = lane & 15;
  float* P = lds[wave];

  // Q tile as A operand (kept in registers for the whole key loop)
  v2f aq[8];
  {
    const float* qrow = Q + (size_t)(row0 + m) * D_;
#pragma unroll
    for (int k = 0; k < 8; ++k) aq[k] = *(const v2f*)(qrow + 4 * k + 2 * h);
  }

  float mrow[8], lrow[8];
#pragma unroll
  for (int r = 0; r < 8; ++r) {
    mrow[r] = -INFINITY;
    lrow[r] = 0.0f;
  }
  v8f o0 = {};  // ctx feats 0..15
  v8f o1 = {};  // ctx feats 16..31

  const float* Kb = Kmat + (size_t)b * S_ * D_;
  const float* Vtb = Vt + (size_t)b * D_ * S_;
  const float* vt0 = Vtb + (size_t)m * S_;         // feat row m   (N-tile 0)
  const float* vt1 = Vtb + (size_t)(16 + m) * S_;  // feat row 16+m (N-tile 1)

  for (int kt = 0; kt < S_ / 32; ++kt) {
    // ---- scores: two 16x16 tiles, S = Q[16x32] @ K_tile^T[32x32] ----------
    const float* krow0 = Kb + (size_t)(kt * 32 + m) * D_;   // keys kt*32+0..15
    const float* krow1 = krow0 + 16 * D_;                   // keys kt*32+16..31
    v8f s0 = {};
    v8f s1 = {};
#pragma unroll
    for (int k = 0; k < 8; ++k) {
      v2f b0 = *(const v2f*)(krow0 + 4 * k + 2 * h);
      v2f b1 = *(const v2f*)(krow1 + 4 * k + 2 * h);
      s0 = wmma_f32(aq[k], b0, s0);
      s1 = wmma_f32(aq[k], b1, s1);
    }
    // Prefetch next iteration's K rows and V columns into the caches.
    __builtin_prefetch(krow0 + 32 * D_, 0, 3);
    __builtin_prefetch(krow1 + 32 * D_, 0, 3);
    __builtin_prefetch(vt0 + kt * 32 + 32, 0, 3);
    __builtin_prefetch(vt1 + kt * 32 + 32, 0, 3);

    // ---- online softmax update over 32 keys (rows live in 16-lane halves) -
#pragma unroll
    for (int r = 0; r < 8; ++r) {
      float v = fmaxf(s0[r], s1[r]);
      v = fmaxf(v, __shfl_xor(v, 1));
      v = fmaxf(v, __shfl_xor(v, 2));
      v = fmaxf(v, __shfl_xor(v, 4));
      v = fmaxf(v, __shfl_xor(v, 8));
      const float mnew = fmaxf(mrow[r], v);
      const float alpha = __expf(mrow[r] - mnew);
      mrow[r] = mnew;
      const float p0 = __expf(s0[r] - mnew);
      const float p1 = __expf(s1[r] - mnew);
      s0[r] = p0;
      s1[r] = p1;
      float ps = p0 + p1;
      ps += __shfl_xor(ps, 1);
      ps += __shfl_xor(ps, 2);
      ps += __shfl_xor(ps, 4);
      ps += __shfl_xor(ps, 8);
      lrow[r] = lrow[r] * alpha + ps;
      o0[r] *= alpha;
      o1[r] *= alpha;
    }
    // ---- P tile [16 x 32]: C/D layout -> row-major LDS --------------------
#pragma unroll
    for (int r = 0; r < 8; ++r) {
      P[(r + 8 * h) * 32 + m] = s0[r];
      P[(r + 8 * h) * 32 + 16 + m] = s1[r];
    }
    // ---- O += P[16x32] @ V_tile[32x32] (V from transposed layout, b64) ----
#pragma unroll
    for (int k = 0; k < 8; ++k) {
      v2f ap = *(const v2f*)(&P[m * 32 + 4 * k + 2 * h]);
      v2f bv0 = *(const v2f*)(vt0 + kt * 32 + 4 * k + 2 * h);
      v2f bv1 = *(const v2f*)(vt1 + kt * 32 + 4 * k + 2 * h);
      o0 = wmma_f32(ap, bv0, o0);
      o1 = wmma_f32(ap, bv1, o1);
    }
  }

  // ---- normalize by softmax denominator -----------------------------------
#pragma unroll
  for (int r = 0; r < 8; ++r) {
    const float inv = 1.0f / lrow[r];
    o0[r] *= inv;
    o1[r] *= inv;
  }
  // ---- ctx tile -> LDS (16x32 row-major) -> A operand ----------------------
#pragma unroll
  for (int r = 0; r < 8; ++r) {
    P[(r + 8 * h) * 32 + m] = o0[r];
    P[(r + 8 * h) * 32 + 16 + m] = o1[r];
  }
  v2f ac[8];
#pragma unroll
  for (int k = 0; k < 8; ++k) ac[k] = *(const v2f*)(&P[m * 32 + 4 * k + 2 * h]);

  // ---- out = tanh(ctx @ Wf + bf) ------------------------------------------
#pragma unroll
  for (int nt = 0; nt < 2; ++nt) {
    const int n = nt * 16 + m;
    v8f c;
    const float bb0 = bf[n];
#pragma unroll
    for (int r = 0; r < 8; ++r) c[r] = bb0;
#pragma unroll
    for (int k = 0; k < 8; ++k) {
      const int f = 4 * k + 2 * h;
      v2f bb;
      bb.x = Wf[(f + 0) * D_ + n];
      bb.y = Wf[(f + 1) * D_ + n];
      c = wmma_f32(ac[k], bb, c);
    }
#pragma unroll
    for (int r = 0; r < 8; ++r)
      out[(size_t)(row0 + r + 8 * h) * D_ + n] = tanhf(c[r]);
  }
}

// ---------------------------------------------------------------------------
extern "C" void kernel_launch(void* const* d_in, const int* in_sizes, int n_in,
                              void* d_out, int out_size, void* d_ws,
                              size_t ws_size, hipStream_t stream) {
  const float* X = (const float*)d_in[0];
  const float* Wq = (const float*)d_in[1];
  const float* bq = (const float*)d_in[2];
  const float* Wk = (const float*)d_in[3];
  const float* bk = (const float*)d_in[4];
  const float* Wv = (const float*)d_in[5];
  const float* bv = (const float*)d_in[6];
  const float* Wf = (const float*)d_in[7];
  const float* bf = (const float*)d_in[8];

  // Workspace: Q, K row-major; V transposed [B][D][S]  (3 x 2 MB)
  float* Q = (float*)d_ws;
  float* K = Q + (size_t)B_ * S_ * D_;
  float* Vt = K + (size_t)B_ * S_ * D_;

  const int tiles = (B_ * S_) / 16;  // 1024 waves, 8 waves/block
  qkv_proj_kernel<<<tiles / 8, 256, 0, stream>>>(X, Wq, bq, Wk, bk, Wv, bv, Q,
                                                 K, Vt);
  attn_kernel<<<tiles / 8, 256, 0, stream>>>(Q, K, Vt, Wf, bf, (float*)d_out);
}